// FastCausalSetAttention_50861002719449
// MI455X (gfx1250) — compile-verified
//
#include <hip/hip_runtime.h>
#include <hip/hip_bf16.h>

// ---------------------------------------------------------------------------
// FastCausalSetAttention on gfx1250 (MI455X), fp32 end-to-end via
// V_WMMA_F32_16X16X4_F32.  Shapes fixed: B=1, T=1024, C=512, NH=16, HS=32,
// 4 sets with L=[0,.5,0,.75], R=[1,1,.5,1].
//
// qkv scratch layout (remapped by GEMM1 store):
//   chunk = col/32  (0..15 q heads | 16..31 k heads | 32..47 v heads)
//   offset = (chunk*T + t)*32 + (col&31)   -> per-head rows of 32 contiguous.
// ---------------------------------------------------------------------------

typedef __attribute__((ext_vector_type(2))) float v2f;
typedef __attribute__((ext_vector_type(8))) float v8f;

#define T_DIM   1024
#define C_DIM   512
#define NH_DIM  16
#define HS_DIM  32
#define QKV_COLS 1536

__device__ __forceinline__ v8f wmma_f32x4(v2f a, v2f b, v8f c) {
  // D(16x16,f32) = A(16x4,f32) x B(4x16,f32) + C
  return __builtin_amdgcn_wmma_f32_16x16x4_f32(false, a, false, b, (short)0, c,
                                               false, false);
}

// ---------------------------------------------------------------------------
// C = A[M,K] * B[N,K]^T, fp32 row-major inputs.  One wave: 32x64 strip,
// 8 accumulators, register double-buffered K chunks of 4.
// remap!=0 stores C into the per-head qkv layout described above.
// Requires M%32==0, N%64==0, K%8==0.
// ---------------------------------------------------------------------------
__global__ __launch_bounds__(256) void gemm_nt_wmma_f32(
    const float* __restrict__ A, const float* __restrict__ B,
    float* __restrict__ C, int M, int N, int K, int remap) {
  const int wave = threadIdx.x >> 5;
  const int lane = threadIdx.x & 31;
  const int half = lane >> 4;     // K pair {0,1} vs {2,3}
  const int idx  = lane & 15;     // M (A) or N (B) within 16-tile

  const int ntiles = N >> 6;
  const int wid = blockIdx.x * (int)(blockDim.x >> 5) + wave;
  const int mt = wid / ntiles;
  const int nt = wid % ntiles;
  if (mt * 32 >= M) return;       // uniform per wave

  const int m0 = mt * 32;
  const int n0 = nt * 64;
  const int o  = half * 2;

  const float* __restrict__ ar0 = A + (size_t)(m0 + idx) * K + o;
  const float* __restrict__ ar1 = A + (size_t)(m0 + 16 + idx) * K + o;
  const float* __restrict__ br  = B + (size_t)(n0 + idx) * K + o;

  v8f acc[2][4];
#pragma unroll
  for (int mi = 0; mi < 2; ++mi)
#pragma unroll
    for (int j = 0; j < 4; ++j) acc[mi][j] = (v8f){};

  v2f a0[2], b0[4], a1[2], b1[4];

#define LOAD_SET(av, bv, k0)                                                  \
  do {                                                                        \
    av[0] = *(const v2f*)(ar0 + (k0));                                        \
    av[1] = *(const v2f*)(ar1 + (k0));                                        \
    bv[0] = *(const v2f*)(br + (k0));                                         \
    bv[1] = *(const v2f*)(br + (size_t)16 * K + (k0));                        \
    bv[2] = *(const v2f*)(br + (size_t)32 * K + (k0));                        \
    bv[3] = *(const v2f*)(br + (size_t)48 * K + (k0));                        \
  } while (0)

#define WMMA_SET(av, bv)                                                      \
  do {                                                                        \
    _Pragma("unroll") for (int mi = 0; mi < 2; ++mi)                          \
        _Pragma("unroll") for (int j = 0; j < 4; ++j)                         \
            acc[mi][j] = wmma_f32x4(av[mi], bv[j], acc[mi][j]);               \
  } while (0)

  LOAD_SET(a0, b0, 0);
  for (int k0 = 0; k0 < K - 8; k0 += 8) {
    LOAD_SET(a1, b1, k0 + 4);   // prefetch while consuming a0/b0
    WMMA_SET(a0, b0);
    LOAD_SET(a0, b0, k0 + 8);   // prefetch while consuming a1/b1
    WMMA_SET(a1, b1);
  }
  LOAD_SET(a1, b1, K - 4);      // chunk K-8 already in a0/b0
  WMMA_SET(a0, b0);
  WMMA_SET(a1, b1);
#undef LOAD_SET
#undef WMMA_SET

  // D layout: lane idx = col-within-16; VGPR v = row v + 8*half.
#pragma unroll
  for (int mi = 0; mi < 2; ++mi)
#pragma unroll
    for (int v = 0; v < 8; ++v) {
      const int m = m0 + mi * 16 + v + 8 * half;
#pragma unroll
      for (int j = 0; j < 4; ++j) {
        const int n = n0 + j * 16 + idx;
        const size_t off = remap
            ? ((size_t)(n >> 5) * T_DIM + m) * HS_DIM + (n & 31)
            : (size_t)m * N + n;
        C[off] = acc[mi][j][v];
      }
    }
}

// ---------------------------------------------------------------------------
// Set-attention core.  One wave per (head h, 16-query tile qt).
// qkv in per-head layout: q at chunk h, k at chunk 16+h, v at chunk 32+h,
// each (T x 32) row-major.  Output attn_out (T x C) row-major.
// ---------------------------------------------------------------------------
__global__ __launch_bounds__(256) void set_attn_wmma(
    const float* __restrict__ qkv, float* __restrict__ aout) {
  __shared__ float lds[8][16 * 18];   // per-wave 16x16 transpose tile, pad 18

  const int wave = threadIdx.x >> 5;
  const int lane = threadIdx.x & 31;
  const int half = lane >> 4;
  const int idx  = lane & 15;

  const int w  = blockIdx.x * 8 + wave;    // 1024 waves
  const int h  = w & (NH_DIM - 1);
  const int qt = w >> 4;                   // uniform across block
  const int t0 = qt * 16;

  const float* __restrict__ qbase = qkv + (size_t)h * T_DIM * HS_DIM;
  const float* __restrict__ kbase = qkv + (size_t)(16 + h) * T_DIM * HS_DIM;
  const float* __restrict__ vbase = qkv + (size_t)(32 + h) * T_DIM * HS_DIM;

  // ---- Q fragments (A-layout): lane idx = m, 8 K-chunks of 4 over hs=32 ---
  v2f qa[8];
  {
    const float* qrow = qbase + (size_t)(t0 + idx) * HS_DIM + half * 2;
#pragma unroll
    for (int c = 0; c < 8; ++c) qa[c] = *(const v2f*)(qrow + c * 4);
  }

  // ---- per-row set bounds (row m = v + 8*half, t = t0 + m) ----------------
  int lb[8][4], rb[8][4];
#pragma unroll
  for (int v = 0; v < 8; ++v) {
    const float tf = (float)(t0 + v + 8 * half);
    const float Ls[4] = {0.0f, 0.5f, 0.0f, 0.75f};
    const float Rs[4] = {1.0f, 1.0f, 0.5f, 1.0f};
#pragma unroll
    for (int s = 0; s < 4; ++s) {
      lb[v][s] = (int)fmaxf(tf * Ls[s] - 1.0f, -1.0f);  // trunc-toward-0 cast
      rb[v][s] = (int)floorf(tf * Rs[s]);
    }
  }

  const int jmax = qt;

  auto loadK = [&](v2f (&kb)[8], int j) {
    const float* krow = kbase + (size_t)(j * 16 + idx) * HS_DIM + half * 2;
#pragma unroll
    for (int c = 0; c < 8; ++c) kb[c] = *(const v2f*)(krow + c * 4);
  };
  auto scoreTile = [&](const v2f (&kb)[8]) {
    v8f sc = {};
#pragma unroll
    for (int c = 0; c < 8; ++c) sc = wmma_f32x4(qa[c], kb[c], sc);
    return sc;
  };

  // ---- pass 1: per-set range sums of scores, double-buffered K tiles ------
  float asum[4][8];
#pragma unroll
  for (int s = 0; s < 4; ++s)
#pragma unroll
    for (int v = 0; v < 8; ++v) asum[s][v] = 0.0f;

  auto accumulate = [&](const v8f& sc, int j) {
    const int tp = j * 16 + idx;  // this lane's key index n
#pragma unroll
    for (int v = 0; v < 8; ++v) {
      const float sv = sc[v];
#pragma unroll
      for (int s = 0; s < 4; ++s)
        if (tp > lb[v][s] && tp <= rb[v][s]) asum[s][v] += sv;
    }
  };

  {
    v2f kb0[8], kb1[8];
    loadK(kb0, 0);
    for (int j = 0;;) {
      if (j < jmax) loadK(kb1, j + 1);
      accumulate(scoreTile(kb0), j);
      if (++j > jmax) break;
      if (j < jmax) loadK(kb0, j + 1);
      accumulate(scoreTile(kb1), j);
      if (++j > jmax) break;
    }
  }

  // reduce over the 16 lanes sharing each row (xor tree stays in-half)
#pragma unroll
  for (int s = 0; s < 4; ++s)
#pragma unroll
    for (int v = 0; v < 8; ++v) {
      float x = asum[s][v];
      x += __shfl_xor(x, 1, 32);
      x += __shfl_xor(x, 2, 32);
      x += __shfl_xor(x, 4, 32);
      x += __shfl_xor(x, 8, 32);
      asum[s][v] = x;
    }

  // ---- softmax over the 4 sets (register-resident) ------------------------
  float att[4][8];
#pragma unroll
  for (int v = 0; v < 8; ++v) {
    const float mx = fmaxf(fmaxf(asum[0][v], asum[1][v]),
                           fmaxf(asum[2][v], asum[3][v]));
    const float e0 = __expf(asum[0][v] - mx);
    const float e1 = __expf(asum[1][v] - mx);
    const float e2 = __expf(asum[2][v] - mx);
    const float e3 = __expf(asum[3][v] - mx);
    const float inv = 1.0f / (e0 + e1 + e2 + e3);
    att[0][v] = e0 * inv; att[1][v] = e1 * inv;
    att[2][v] = e2 * inv; att[3][v] = e3 * inv;
  }

  // ---- pass 2: out = (w .* S) @ V -----------------------------------------
  v8f oacc0 = {}, oacc1 = {};
  float* __restrict__ my = lds[wave];

  for (int j = 0; j <= jmax; ++j) {
    v2f kb[8];
    loadK(kb, j);
    // V fragments for this tile: independent of the score WMMAs -> the
    // compiler can overlap these loads with the WMMA chain below.
    v2f vb[4][2];
    {
      const float* vrow = vbase + (size_t)(j * 16 + half * 2) * HS_DIM + idx;
#pragma unroll
      for (int c = 0; c < 4; ++c)
#pragma unroll
        for (int eh = 0; eh < 2; ++eh) {
          vb[c][eh].x = vrow[(c * 4) * HS_DIM + eh * 16];
          vb[c][eh].y = vrow[(c * 4 + 1) * HS_DIM + eh * 16];
        }
    }

    v8f sc = scoreTile(kb);

    const int tp = j * 16 + idx;
#pragma unroll
    for (int v = 0; v < 8; ++v) {
      float wsum = 0.0f;
#pragma unroll
      for (int s = 0; s < 4; ++s)
        if (tp > lb[v][s] && tp <= rb[v][s]) wsum += att[s][v];
      my[(v + 8 * half) * 18 + idx] = sc[v] * wsum;  // D-layout -> LDS tile
    }
    // same-wave LDS ops are in-order; fence compiler + CDNA5 DS counter
    asm volatile("s_wait_dscnt 0" ::: "memory");

    // P as A-fragments (transpose read: lane idx = m, K-chunks over n)
    v2f pa[4];
#pragma unroll
    for (int c = 0; c < 4; ++c)
      pa[c] = *(const v2f*)(my + idx * 18 + c * 4 + half * 2);

#pragma unroll
    for (int c = 0; c < 4; ++c) {
      oacc0 = wmma_f32x4(pa[c], vb[c][0], oacc0);
      oacc1 = wmma_f32x4(pa[c], vb[c][1], oacc1);
    }
  }

  // ---- store attn_out (T x C): row t, cols h*32 + e -----------------------
#pragma unroll
  for (int v = 0; v < 8; ++v) {
    float* orow = aout + (size_t)(t0 + v + 8 * half) * C_DIM + h * HS_DIM + idx;
    orow[0]  = oacc0[v];
    orow[16] = oacc1[v];
  }
}

// ---------------------------------------------------------------------------
// Launcher.  d_in: [x (1,1024,512) f32, W_attn (1536,512) f32,
// W_proj (512,512) f32].  d_out: (1,1024,512) f32.
// ---------------------------------------------------------------------------
extern "C" void kernel_launch(void* const* d_in, const int* in_sizes, int n_in,
                              void* d_out, int out_size, void* d_ws,
                              size_t ws_size, hipStream_t stream) {
  (void)in_sizes; (void)n_in; (void)out_size; (void)ws_size;

  const float* x      = (const float*)d_in[0];
  const float* W_attn = (const float*)d_in[1];
  const float* W_proj = (const float*)d_in[2];
  float* out = (float*)d_out;

  float* qkv  = (float*)d_ws;                        // 48 chunks * 1024 * 32
  float* aout = qkv + (size_t)T_DIM * QKV_COLS;      // 1024 * 512

  // 1) qkv = x @ W_attn^T, stored in per-head layout (remap=1)
  {
    const int waves = (T_DIM / 32) * (QKV_COLS / 64);   // 768
    gemm_nt_wmma_f32<<<waves / 8, 256, 0, stream>>>(x, W_attn, qkv,
                                                    T_DIM, QKV_COLS, C_DIM, 1);
  }
  // 2) set attention: 16 heads x 64 query tiles = 1024 waves
  set_attn_wmma<<<(NH_DIM * (T_DIM / 16)) / 8, 256, 0, stream>>>(qkv, aout);

  // 3) out = aout @ W_proj^T (plain row-major store)
  {
    const int waves = (T_DIM / 32) * (C_DIM / 64);      // 256
    gemm_nt_wmma_f32<<<waves / 8, 256, 0, stream>>>(aout, W_proj, out,
                                                    T_DIM, C_DIM, C_DIM, 0);
  }
}